// StagedGGNN_26912265076896
// MI455X (gfx1250) — compile-verified
//
#include <hip/hip_runtime.h>
#include <math.h>

typedef __attribute__((ext_vector_type(2))) float v2f;
typedef __attribute__((ext_vector_type(8))) float v8f;

#define GGNN_STEPS 3
#define WMMA_F32(A, B, C) \
    __builtin_amdgcn_wmma_f32_16x16x4_f32(false, (A), false, (B), (short)0, (C), false, false)

// ---------------------------------------------------------------------------
// WMMA fp32 GEMM, Kdim fixed at 128, Ncols multiple of 128:
//   Out[M x Ncols] = X[M x 128] @ W[Ncols x 128]^T  (+bias | +=)
// Block = 256 threads (8 waves) -> 128x128 output tile.
// Both operand tiles are staged in LDS via GLOBAL_LOAD_ASYNC_TO_LDS_B128
// (ASYNCcnt), then the inner loop is pure ds_load_b64 + v_wmma (no vmem waits).
// Wave tile: 32 rows x 64 cols = 8 v8f accumulators; 6 LDS loads / 8 WMMAs.
// LDS rows padded to 132 floats: keeps 16B alignment for b128 async stores and
// spreads the 16 lane addresses across banks for ds_load_b64.
// ---------------------------------------------------------------------------
#define LDSTRIDE 132

__global__ __launch_bounds__(256)
void gemm_nt_wmma(const float* __restrict__ X, int ldx,
                  const float* __restrict__ W, int ldw,
                  const float* __restrict__ bias,
                  float* __restrict__ Out, int ldo,
                  int M, int Ncols, int accumulate)
{
    __shared__ __align__(16) float Ws[128 * LDSTRIDE];
    __shared__ __align__(16) float Xs[128 * LDSTRIDE];

    const int tid  = threadIdx.x;
    const int lane = tid & 31;
    const int wave = tid >> 5;
    const int l16  = lane & 15;
    const int half = lane >> 4;     // 0: K=0..1 / rows 0..7 ; 1: K=2..3 / rows 8..15

    const int rowBlk = blockIdx.x * 128;
    const int colBlk = blockIdx.y * 128;

    // ---- async-stage W[colBlk..+127][0..127] and X[rowBlk..+127][0..127] ----
    // 128 rows x 32 b128-chunks per matrix; 16 chunks per thread per matrix.
    for (int c = tid; c < 128 * 32; c += 256) {
        const int r  = c >> 5;
        const int kc = (c & 31) * 4;                      // float index of 16B chunk
        const float* gw = W + (size_t)(colBlk + r) * ldw + kc;
        const unsigned lw = (unsigned)(size_t)&Ws[r * LDSTRIDE + kc];
        asm volatile("global_load_async_to_lds_b128 %0, %1, off"
                     :: "v"(lw), "v"(gw) : "memory");
        const int xr = min(rowBlk + r, M - 1);            // clamp ragged M
        const float* gx = X + (size_t)xr * ldx + kc;
        const unsigned lx = (unsigned)(size_t)&Xs[r * LDSTRIDE + kc];
        asm volatile("global_load_async_to_lds_b128 %0, %1, off"
                     :: "v"(lx), "v"(gx) : "memory");
    }
    asm volatile("s_wait_asynccnt 0x0" ::: "memory");
    __syncthreads();

    // ---- wave tile: rows mrow0..+31, cols ncol0..+63 (within block) ----
    const int mrow0 = (wave & 3) * 32;
    const int ncol0 = (wave >> 2) * 64;

    const float* XsA0 = &Xs[(mrow0 +  0 + l16) * LDSTRIDE + 2 * half];
    const float* XsA1 = &Xs[(mrow0 + 16 + l16) * LDSTRIDE + 2 * half];
    const float* WsB0 = &Ws[(ncol0 +  0 + l16) * LDSTRIDE + 2 * half];
    const float* WsB1 = &Ws[(ncol0 + 16 + l16) * LDSTRIDE + 2 * half];
    const float* WsB2 = &Ws[(ncol0 + 32 + l16) * LDSTRIDE + 2 * half];
    const float* WsB3 = &Ws[(ncol0 + 48 + l16) * LDSTRIDE + 2 * half];

    v8f a00 = {}, a01 = {}, a02 = {}, a03 = {};
    v8f a10 = {}, a11 = {}, a12 = {}, a13 = {};

    #pragma unroll 4
    for (int kk = 0; kk < 128; kk += 4) {
        const v2f va0 = *(const v2f*)(XsA0 + kk);
        const v2f va1 = *(const v2f*)(XsA1 + kk);
        const v2f vb0 = *(const v2f*)(WsB0 + kk);
        const v2f vb1 = *(const v2f*)(WsB1 + kk);
        const v2f vb2 = *(const v2f*)(WsB2 + kk);
        const v2f vb3 = *(const v2f*)(WsB3 + kk);
        a00 = WMMA_F32(va0, vb0, a00);
        a01 = WMMA_F32(va0, vb1, a01);
        a02 = WMMA_F32(va0, vb2, a02);
        a03 = WMMA_F32(va0, vb3, a03);
        a10 = WMMA_F32(va1, vb0, a10);
        a11 = WMMA_F32(va1, vb1, a11);
        a12 = WMMA_F32(va1, vb2, a12);
        a13 = WMMA_F32(va1, vb3, a13);
    }

    // ---- epilogue: C/D layout = VGPR v -> M=v (lanes 0-15) / M=v+8 (16-31) ----
#define STORE_ACC(ACC, MT, NT)                                                  \
    {                                                                           \
        const int col = colBlk + ncol0 + (NT) * 16 + l16;                       \
        const float bv = (bias != nullptr) ? bias[col] : 0.0f;                  \
        _Pragma("unroll")                                                       \
        for (int v = 0; v < 8; ++v) {                                           \
            const int row = rowBlk + mrow0 + (MT) * 16 + v + 8 * half;          \
            if (row < M) {                                                      \
                const size_t o = (size_t)row * ldo + col;                       \
                if (accumulate) Out[o] += (ACC)[v];                             \
                else            Out[o] = (ACC)[v] + bv;                         \
            }                                                                   \
        }                                                                       \
    }
    STORE_ACC(a00, 0, 0) STORE_ACC(a01, 0, 1) STORE_ACC(a02, 0, 2) STORE_ACC(a03, 0, 3)
    STORE_ACC(a10, 1, 0) STORE_ACC(a11, 1, 1) STORE_ACC(a12, 1, 2) STORE_ACC(a13, 1, 3)
#undef STORE_ACC
}

// ---------------------------------------------------------------------------
// Edge pass for a single etype k: a[dst] += t_k[src] for edges of type k.
// One wave per edge; src/dst/etype forced scalar (SMEM) via readfirstlane.
// Gather source t_k (51 MB) and accumulator a (51 MB) are both L2-resident.
// ---------------------------------------------------------------------------
__global__ __launch_bounds__(256)
void edge_scatter(const float* __restrict__ t_k,
                  const int* __restrict__ src, const int* __restrict__ dst,
                  const int* __restrict__ etypes, int k,
                  float* __restrict__ a, int E)
{
    const int lane = threadIdx.x & 31;
    int e = (blockIdx.x * 256 + threadIdx.x) >> 5;       // wave-uniform
    e = __builtin_amdgcn_readfirstlane(e);
    if (e >= E) return;
    if (__builtin_amdgcn_readfirstlane(etypes[e]) != k) return;
    const int s = __builtin_amdgcn_readfirstlane(src[e]);
    const int d = __builtin_amdgcn_readfirstlane(dst[e]);
    const float4 v = ((const float4*)(t_k + (size_t)s * 128))[lane];
    float* ap = a + (size_t)d * 128 + lane * 4;
    unsafeAtomicAdd(ap + 0, v.x);
    unsafeAtomicAdd(ap + 1, v.y);
    unsafeAtomicAdd(ap + 2, v.z);
    unsafeAtomicAdd(ap + 3, v.w);
}

__global__ __launch_bounds__(256)
void zero_f32(float4* __restrict__ p, int n4)
{
    const int i = blockIdx.x * 256 + threadIdx.x;
    if (i < n4) p[i] = make_float4(0.f, 0.f, 0.f, 0.f);
}

__device__ __forceinline__ float sigmf(float x) { return 1.0f / (1.0f + __expf(-x)); }

// GRU elementwise, torch gate order (r, z, n), float4 vectorized.
__global__ __launch_bounds__(256)
void gru_elem(const float4* __restrict__ gi, const float4* __restrict__ gh,
              const float4* __restrict__ h, float4* __restrict__ h_out, int N)
{
    const int i = blockIdx.x * 256 + threadIdx.x;        // over N*32
    if (i >= N * 32) return;
    const int row = i >> 5, c4 = i & 31;
    const size_t b = (size_t)row * 96;                   // 384 floats = 96 float4
    const float4 giR = gi[b + c4],      ghR = gh[b + c4];
    const float4 giZ = gi[b + 32 + c4], ghZ = gh[b + 32 + c4];
    const float4 giN = gi[b + 64 + c4], ghN = gh[b + 64 + c4];
    const float4 hv  = h[i];
    float4 o;
    {
        const float r = sigmf(giR.x + ghR.x), z = sigmf(giZ.x + ghZ.x);
        const float n = tanhf(giN.x + r * ghN.x);
        o.x = (1.0f - z) * n + z * hv.x;
    }
    {
        const float r = sigmf(giR.y + ghR.y), z = sigmf(giZ.y + ghZ.y);
        const float n = tanhf(giN.y + r * ghN.y);
        o.y = (1.0f - z) * n + z * hv.y;
    }
    {
        const float r = sigmf(giR.z + ghR.z), z = sigmf(giZ.z + ghZ.z);
        const float n = tanhf(giN.z + r * ghN.z);
        o.z = (1.0f - z) * n + z * hv.z;
    }
    {
        const float r = sigmf(giR.w + ghR.w), z = sigmf(giZ.w + ghZ.w);
        const float n = tanhf(giN.w + r * ghN.w);
        o.w = (1.0f - z) * n + z * hv.w;
    }
    h_out[i] = o;
}

__global__ __launch_bounds__(256)
void head_elem(const float4* __restrict__ gpre, const float4* __restrict__ jpre,
               float4* __restrict__ out, int n4)
{
    const int i = blockIdx.x * 256 + threadIdx.x;
    if (i >= n4) return;
    const float4 g = gpre[i], j = jpre[i];
    out[i] = make_float4(sigmf(g.x) * j.x, sigmf(g.y) * j.y,
                         sigmf(g.z) * j.z, sigmf(g.w) * j.w);
}

// ---------------------------------------------------------------------------
extern "C" void kernel_launch(void* const* d_in, const int* in_sizes, int n_in,
                              void* d_out, int out_size, void* d_ws, size_t ws_size,
                              hipStream_t stream)
{
    const float* features = (const float*)d_in[0];
    const int*   src      = (const int*)  d_in[1];
    const int*   dst      = (const int*)  d_in[2];
    const int*   etypes   = (const int*)  d_in[3];
    const float* W_et     = (const float*)d_in[4];
    const float* b_et     = (const float*)d_in[5];
    const float* W_ih     = (const float*)d_in[6];
    const float* b_ih     = (const float*)d_in[7];
    const float* W_hh     = (const float*)d_in[8];
    const float* b_hh     = (const float*)d_in[9];
    const float* i_w      = (const float*)d_in[10];
    const float* i_b      = (const float*)d_in[11];
    const float* j_w      = (const float*)d_in[12];
    const float* j_b      = (const float*)d_in[13];

    const int D = 128, K = 4;
    const int N = in_sizes[0] / D;
    const int E = in_sizes[1];
    const size_t nD = (size_t)N * D;
    const int nD4 = (int)(nD / 4);

    // Workspace (floats): h0 | h1 | a | bufD(N*384) | bufE(N*384)
    // bufD doubles as t_k and head-gate preact; bufE as gh / head-j preact.
    float* ws   = (float*)d_ws;
    float* h0   = ws;
    float* h1   = h0 + nD;
    float* abuf = h1 + nD;
    float* bufD = abuf + nD;
    float* bufE = bufD + (size_t)N * 384;

    const dim3 blk(256);
    auto gemm_grid = [](int M, int Nc) { return dim3((M + 127) / 128, Nc / 128); };
    const int ew_blocks = (int)(((size_t)E * 32 + 255) / 256);
    const int v4_blocks = (nD4 + 255) / 256;

    const float* h_in = features;
    float* h_out = h0;
    for (int s = 0; s < GGNN_STEPS; ++s) {
        zero_f32<<<v4_blocks, blk, 0, stream>>>((float4*)abuf, nD4);
        // One etype at a time: keeps t_k (51 MB) and a (51 MB) L2-resident
        for (int k = 0; k < K; ++k) {
            gemm_nt_wmma<<<gemm_grid(N, D), blk, 0, stream>>>(
                h_in, D, W_et + (size_t)k * D * D, D, b_et + k * D,
                bufD, D, N, D, 0);
            edge_scatter<<<ew_blocks, blk, 0, stream>>>(bufD, src, dst, etypes, k, abuf, E);
        }
        // GRU gates: gi = a @ W_ih^T + b_ih ; gh = h @ W_hh^T + b_hh
        gemm_nt_wmma<<<gemm_grid(N, 3 * D), blk, 0, stream>>>(
            abuf, D, W_ih, D, b_ih, bufD, 3 * D, N, 3 * D, 0);
        gemm_nt_wmma<<<gemm_grid(N, 3 * D), blk, 0, stream>>>(
            h_in, D, W_hh, D, b_hh, bufE, 3 * D, N, 3 * D, 0);
        gru_elem<<<(N * 32 + 255) / 256, blk, 0, stream>>>(
            (const float4*)bufD, (const float4*)bufE, (const float4*)h_in,
            (float4*)h_out, N);
        h_in = h_out;
        h_out = (h_out == h0) ? h1 : h0;
    }

    // Gated head: gate = sig([h, features] @ i_w^T + i_b); out = gate * (h @ j_w^T + j_b)
    gemm_nt_wmma<<<gemm_grid(N, D), blk, 0, stream>>>(
        h_in, D, i_w, 2 * D, i_b, bufD, D, N, D, 0);                // h vs left half of i_w
    gemm_nt_wmma<<<gemm_grid(N, D), blk, 0, stream>>>(
        features, D, i_w + D, 2 * D, nullptr, bufD, D, N, D, 1);    // += features vs right half
    gemm_nt_wmma<<<gemm_grid(N, D), blk, 0, stream>>>(
        h_in, D, j_w, D, j_b, bufE, D, N, D, 0);
    head_elem<<<v4_blocks, blk, 0, stream>>>(
        (const float4*)bufD, (const float4*)bufE, (float4*)d_out, nD4);
}